// EdgesConvLayer_75144747811216
// MI455X (gfx1250) — compile-verified
//
#include <hip/hip_runtime.h>

#define N_NODES 32768
#define N_EDGES 262144
#define C_DIM   128
#define H_HEADS 8
#define DKH     16
#define R_REL   8

typedef __attribute__((ext_vector_type(2))) float v2f;
typedef __attribute__((ext_vector_type(8))) float v8f;
typedef __attribute__((__vector_size__(4 * sizeof(int)))) int i32x4;

// ---- order-preserving float <-> uint encoding (for atomic segment-max) ----
__device__ __forceinline__ unsigned enc_f32(float f) {
  unsigned u = __float_as_uint(f);
  return (u & 0x80000000u) ? ~u : (u | 0x80000000u);
}
__device__ __forceinline__ float dec_f32(unsigned e) {
  unsigned u = (e & 0x80000000u) ? (e & 0x7FFFFFFFu) : ~e;
  return __uint_as_float(u);
}
#define ENC_NEG_INF 0x007FFFFFu   // enc(-inf)

// ---- async bulk copy global -> LDS (ASYNCcnt path), with sync fallback ----
__device__ __forceinline__ void stage_64k_async(const float* __restrict__ gsrc,
                                                float* lds_dst, int tidx) {
#if __has_builtin(__builtin_amdgcn_global_load_async_to_lds_b128)
  auto g = (__attribute__((address_space(1))) i32x4*)gsrc;
  auto l = (__attribute__((address_space(3))) i32x4*)lds_dst;
  for (int i = tidx; i < C_DIM * C_DIM / 4; i += 256)
    __builtin_amdgcn_global_load_async_to_lds_b128(g + i, l + i, 0, 0);
#else
  const float4* g4 = (const float4*)gsrc;
  float4* l4 = (float4*)lds_dst;
  for (int i = tidx; i < C_DIM * C_DIM / 4; i += 256) l4[i] = g4[i];
#endif
}
__device__ __forceinline__ void stage_wait() {
#if __has_builtin(__builtin_amdgcn_global_load_async_to_lds_b128)
#if __has_builtin(__builtin_amdgcn_s_wait_asynccnt)
  __builtin_amdgcn_s_wait_asynccnt(0);
#else
  asm volatile("s_wait_asynccnt 0x0" ::: "memory");
#endif
#endif
}

// --------------------------------------------------------------------------
// Fold the per-relation per-head [16x16] transform into the projection
// weight:  Wc[r][i][h*16+j] = sum_d W[r][i][h*16+d] * rel[r][h][d][j]
// --------------------------------------------------------------------------
__global__ void prep_weights(const float* __restrict__ W,    // [R,128,128]
                             const float* __restrict__ b,    // [R,128]
                             const float* __restrict__ rel,  // [R,H,16,16]
                             float* __restrict__ Wc,         // [R,128,128]
                             float* __restrict__ bc) {       // [R,128]
  const int rh = blockIdx.x;
  const int r = rh >> 3, h = rh & 7;
  const float* Wr    = W   + (size_t)r * C_DIM * C_DIM;
  const float* relrh = rel + (size_t)rh * DKH * DKH;
  for (int e = threadIdx.x; e < C_DIM * DKH; e += blockDim.x) {
    const int i = e >> 4, j = e & 15;
    float acc = 0.f;
    #pragma unroll
    for (int d = 0; d < DKH; ++d)
      acc += Wr[i * C_DIM + h * DKH + d] * relrh[d * DKH + j];
    Wc[(size_t)r * C_DIM * C_DIM + i * C_DIM + h * DKH + j] = acc;
  }
  if (threadIdx.x < DKH) {
    const int j = threadIdx.x;
    float acc = 0.f;
    #pragma unroll
    for (int d = 0; d < DKH; ++d)
      acc += b[r * C_DIM + h * DKH + d] * relrh[d * DKH + j];
    bc[r * C_DIM + h * DKH + j] = acc;
  }
}

// --------------------------------------------------------------------------
__global__ void init_ws(float* __restrict__ sacc, unsigned* __restrict__ nmax,
                        float* __restrict__ den, float* __restrict__ cnt) {
  const int i = blockIdx.x * blockDim.x + threadIdx.x;
  if (i < N_NODES * C_DIM) sacc[i] = 0.f;
  if (i < N_NODES * H_HEADS) { nmax[i] = ENC_NEG_INF; den[i] = 0.f; }
  if (i < N_NODES) cnt[i] = 0.f;
}

// --------------------------------------------------------------------------
// Dense projection GEMM:  out[r] = x @ W[r] + b[r]  (z selects Q / K' / V').
// Both the 64 KB weight tile AND the 64 KB x row-tile are staged into LDS
// with async-to-LDS copies; the K-loop is then pure ds_load + v_wmma.
// Block: 256 threads (8 waves); wave computes 32x64 via 2x4 WMMA tiles.
// --------------------------------------------------------------------------
__global__ __launch_bounds__(256)
void gemm_qkv(const float* __restrict__ x,
              const float* __restrict__ Wq, const float* __restrict__ bq,
              const float* __restrict__ Wk, const float* __restrict__ bk,
              const float* __restrict__ Wv, const float* __restrict__ bv,
              float* __restrict__ Qo, float* __restrict__ Ko,
              float* __restrict__ Vo) {
  __shared__ float wl[C_DIM * C_DIM];   // 64 KB weight tile
  __shared__ float xl[C_DIM * C_DIM];   // 64 KB x row-tile

  const int r = blockIdx.y;
  const int z = blockIdx.z;
  const float* W    = (z == 0) ? Wq : (z == 1) ? Wk : Wv;
  const float* bias = (z == 0) ? bq : (z == 1) ? bk : bv;
  float*       out  = (z == 0) ? Qo : (z == 1) ? Ko : Vo;
  W    += (size_t)r * C_DIM * C_DIM;
  bias += (size_t)r * C_DIM;
  out  += (size_t)r * N_NODES * C_DIM;

  // async stage W tile and X tile into LDS; drain own ASYNCcnt, then barrier
  stage_64k_async(W, wl, threadIdx.x);
  stage_64k_async(x + (size_t)blockIdx.x * 128 * C_DIM, xl, threadIdx.x);
  stage_wait();
  __syncthreads();

  const int wave = threadIdx.x >> 5;
  const int lane = threadIdx.x & 31;
  const int lm   = lane & 15;
  const int kb   = (lane >> 4) * 2;          // ISA A/B layout: K0/K2 split
  const int waveRow = wave & 3;              // 4 row strips of 32
  const int waveCol = wave >> 2;             // 2 col strips of 64
  const int rowBase = blockIdx.x * 128 + waveRow * 32;
  const int colBase = waveCol * 64;

  v8f acc[2][4];
  const v8f vzero = {0.f, 0.f, 0.f, 0.f, 0.f, 0.f, 0.f, 0.f};
  #pragma unroll
  for (int a = 0; a < 2; ++a)
    #pragma unroll
    for (int c = 0; c < 4; ++c) acc[a][c] = vzero;

  for (int k0 = 0; k0 < C_DIM; k0 += 4) {
    v2f af[2];
    #pragma unroll
    for (int rt = 0; rt < 2; ++rt) {
      const int rowLoc = waveRow * 32 + rt * 16 + lm;
      af[rt] = *(const v2f*)(xl + rowLoc * C_DIM + k0 + kb);   // ds_load_b64
    }
    v2f bf[4];
    #pragma unroll
    for (int ct = 0; ct < 4; ++ct) {
      const int n = colBase + ct * 16 + lm;
      bf[ct].x = wl[(k0 + kb)     * C_DIM + n];
      bf[ct].y = wl[(k0 + kb + 1) * C_DIM + n];
    }
    #pragma unroll
    for (int rt = 0; rt < 2; ++rt)
      #pragma unroll
      for (int ct = 0; ct < 4; ++ct)
        acc[rt][ct] = __builtin_amdgcn_wmma_f32_16x16x4_f32(
            false, af[rt], false, bf[ct], (short)0, acc[rt][ct], false, false);
  }

  // bias add + store (C/D layout: VGPR i -> rows i and i+8; N = lane%16)
  #pragma unroll
  for (int ct = 0; ct < 4; ++ct) {
    const int col = colBase + ct * 16 + lm;
    const float bval = bias[col];
    #pragma unroll
    for (int rt = 0; rt < 2; ++rt) {
      v8f c = acc[rt][ct];
      const int row0 = rowBase + rt * 16 + (lane >> 4) * 8;
      #pragma unroll
      for (int i = 0; i < 8; ++i)
        out[(size_t)(row0 + i) * C_DIM + col] = c[i] + bval;
    }
  }
}

// --------------------------------------------------------------------------
// Per-edge attention logits + atomic segment-max over destination nodes.
// --------------------------------------------------------------------------
__global__ void edge_logits(const int* __restrict__ ei, const int* __restrict__ ea,
                            const float* __restrict__ Q, const float* __restrict__ K,
                            const float* __restrict__ pri,
                            float* __restrict__ att, unsigned* __restrict__ nmax) {
  const int e = blockIdx.x * blockDim.x + threadIdx.x;
  if (e >= N_EDGES) return;
  const int sN = ei[e], dN = ei[N_EDGES + e], r = ea[e];
  const float4* qp = (const float4*)(Q + ((size_t)r * N_NODES + dN) * C_DIM);
  const float4* kp = (const float4*)(K + ((size_t)r * N_NODES + sN) * C_DIM);
  #pragma unroll
  for (int h = 0; h < H_HEADS; ++h) {
    float dot = 0.f;
    #pragma unroll
    for (int i = 0; i < 4; ++i) {
      const float4 q4 = qp[h * 4 + i], k4 = kp[h * 4 + i];
      dot += q4.x * k4.x + q4.y * k4.y + q4.z * k4.z + q4.w * k4.w;
    }
    const float a = dot * pri[r * H_HEADS + h] * 0.25f;   // 1/sqrt(16)
    att[(size_t)e * H_HEADS + h] = a;
    atomicMax(&nmax[dN * H_HEADS + h], enc_f32(a));
  }
}

// --------------------------------------------------------------------------
__global__ void edge_exp(const int* __restrict__ ei,
                         float* __restrict__ att, const unsigned* __restrict__ nmax,
                         float* __restrict__ den, float* __restrict__ cnt) {
  const int e = blockIdx.x * blockDim.x + threadIdx.x;
  if (e >= N_EDGES) return;
  const int dN = ei[N_EDGES + e];
  #pragma unroll
  for (int h = 0; h < H_HEADS; ++h) {
    const float a = att[(size_t)e * H_HEADS + h];
    const float m = dec_f32(nmax[dN * H_HEADS + h]);
    const float ex = __expf(a - m);
    att[(size_t)e * H_HEADS + h] = ex;          // overwrite with exp
    atomicAdd(&den[dN * H_HEADS + h], ex);
  }
  atomicAdd(&cnt[dN], 1.0f);
}

// --------------------------------------------------------------------------
// Weighted message scatter: 8 threads per edge (one per head).
// --------------------------------------------------------------------------
__global__ void edge_msg(const int* __restrict__ ei, const int* __restrict__ ea,
                         const float* __restrict__ V, const float* __restrict__ ex,
                         const float* __restrict__ den, float* __restrict__ sacc) {
  const int t = blockIdx.x * blockDim.x + threadIdx.x;
  if (t >= N_EDGES * H_HEADS) return;
  const int e = t >> 3, h = t & 7;
  const int sN = ei[e], dN = ei[N_EDGES + e], r = ea[e];
  const float a = ex[(size_t)e * H_HEADS + h] / den[dN * H_HEADS + h];
  const float4* vp = (const float4*)(V + ((size_t)r * N_NODES + sN) * C_DIM + h * DKH);
  float* sp = sacc + (size_t)dN * C_DIM + h * DKH;
  #pragma unroll
  for (int i = 0; i < 4; ++i) {
    const float4 v4 = vp[i];
    atomicAdd(sp + i * 4 + 0, a * v4.x);
    atomicAdd(sp + i * 4 + 1, a * v4.y);
    atomicAdd(sp + i * 4 + 2, a * v4.z);
    atomicAdd(sp + i * 4 + 3, a * v4.w);
  }
}

// --------------------------------------------------------------------------
// Scatter-mean, per-type output projection, gated residual, LayerNorm.
// One block (128 threads) per node.
// --------------------------------------------------------------------------
__global__ __launch_bounds__(128)
void finalize(const float* __restrict__ x, const int* __restrict__ tid,
              const float* __restrict__ Wa, const float* __restrict__ ba,
              const float* __restrict__ skip, const float* __restrict__ gamma,
              const float* __restrict__ beta,
              const float* __restrict__ sacc, const float* __restrict__ cnt,
              float* __restrict__ out) {
  __shared__ float sx[C_DIM];
  __shared__ float red[C_DIM];
  const int n = blockIdx.x, c = threadIdx.x;
  const int t = tid[n];
  const float cv = cnt[n];
  const float x0 = (cv > 0.f) ? sacc[(size_t)n * C_DIM + c] / cv : 0.f;
  sx[c] = x0;
  __syncthreads();

  const float* Wt = Wa + (size_t)t * C_DIM * C_DIM;
  float acc = ba[t * C_DIM + c];
  for (int d = 0; d < C_DIM; ++d) acc += sx[d] * Wt[d * C_DIM + c];

  const float alpha = 1.f / (1.f + __expf(-skip[t]));
  const float o = acc * alpha + x[(size_t)n * C_DIM + c] * (1.f - alpha);

  red[c] = o; __syncthreads();
  for (int s = 64; s > 0; s >>= 1) { if (c < s) red[c] += red[c + s]; __syncthreads(); }
  const float mu = red[0] * (1.f / C_DIM);
  __syncthreads();
  const float dm = o - mu;
  red[c] = dm * dm; __syncthreads();
  for (int s = 64; s > 0; s >>= 1) { if (c < s) red[c] += red[c + s]; __syncthreads(); }
  const float var = red[0] * (1.f / C_DIM);
  const float xn = dm * rsqrtf(var + 1e-5f);
  out[(size_t)n * C_DIM + c] = xn * gamma[t * C_DIM + c] + beta[t * C_DIM + c];
}

// --------------------------------------------------------------------------
extern "C" void kernel_launch(void* const* d_in, const int* in_sizes, int n_in,
                              void* d_out, int out_size, void* d_ws, size_t ws_size,
                              hipStream_t stream) {
  const float* x    = (const float*)d_in[0];
  const int*   tid  = (const int*)  d_in[1];
  const int*   ei   = (const int*)  d_in[2];
  const int*   ea   = (const int*)  d_in[3];
  const float* Wk   = (const float*)d_in[4];
  const float* bk   = (const float*)d_in[5];
  const float* Wq   = (const float*)d_in[6];
  const float* bq   = (const float*)d_in[7];
  const float* Wv   = (const float*)d_in[8];
  const float* bv   = (const float*)d_in[9];
  const float* Wa   = (const float*)d_in[10];
  const float* ba   = (const float*)d_in[11];
  const float* pri  = (const float*)d_in[12];
  const float* ratt = (const float*)d_in[13];
  const float* rmsg = (const float*)d_in[14];
  const float* skip = (const float*)d_in[15];
  const float* gam  = (const float*)d_in[16];
  const float* bet  = (const float*)d_in[17];
  float* out = (float*)d_out;

  char* ws = (char*)d_ws;
  size_t off = 0;
  auto alloc = [&](size_t bytes) -> void* {
    void* p = ws + off;
    off = (off + bytes + 255) & ~(size_t)255;
    return p;
  };
  float*    WkA  = (float*)   alloc((size_t)R_REL * C_DIM * C_DIM * 4);
  float*    bkA  = (float*)   alloc((size_t)R_REL * C_DIM * 4);
  float*    WvM  = (float*)   alloc((size_t)R_REL * C_DIM * C_DIM * 4);
  float*    bvM  = (float*)   alloc((size_t)R_REL * C_DIM * 4);
  float*    Qb   = (float*)   alloc((size_t)R_REL * N_NODES * C_DIM * 4);
  float*    Kb   = (float*)   alloc((size_t)R_REL * N_NODES * C_DIM * 4);
  float*    Vb   = (float*)   alloc((size_t)R_REL * N_NODES * C_DIM * 4);
  float*    attb = (float*)   alloc((size_t)N_EDGES * H_HEADS * 4);
  unsigned* nmax = (unsigned*)alloc((size_t)N_NODES * H_HEADS * 4);
  float*    den  = (float*)   alloc((size_t)N_NODES * H_HEADS * 4);
  float*    cntb = (float*)   alloc((size_t)N_NODES * 4);
  float*    sacc = (float*)   alloc((size_t)N_NODES * C_DIM * 4);
  (void)ws_size; (void)in_sizes; (void)n_in; (void)out_size;

  prep_weights<<<dim3(R_REL * H_HEADS), 256, 0, stream>>>(Wk, bk, ratt, WkA, bkA);
  prep_weights<<<dim3(R_REL * H_HEADS), 256, 0, stream>>>(Wv, bv, rmsg, WvM, bvM);
  init_ws<<<(N_NODES * C_DIM + 255) / 256, 256, 0, stream>>>(sacc, nmax, den, cntb);
  gemm_qkv<<<dim3(N_NODES / 128, R_REL, 3), 256, 0, stream>>>(
      x, Wq, bq, WkA, bkA, WvM, bvM, Qb, Kb, Vb);
  edge_logits<<<(N_EDGES + 255) / 256, 256, 0, stream>>>(ei, ea, Qb, Kb, pri, attb, nmax);
  edge_exp<<<(N_EDGES + 255) / 256, 256, 0, stream>>>(ei, attb, nmax, den, cntb);
  edge_msg<<<(N_EDGES * H_HEADS + 255) / 256, 256, 0, stream>>>(ei, ea, Vb, attb, den, sacc);
  finalize<<<N_NODES, 128, 0, stream>>>(x, tid, Wa, ba, skip, gam, bet, sacc, cntb, out);
}